// Seq2Seq_49555332662168
// MI455X (gfx1250) — compile-verified
//
#include <hip/hip_runtime.h>
#include <math.h>

typedef __attribute__((ext_vector_type(2))) float v2f;
typedef __attribute__((ext_vector_type(8))) float v8f;

#define HID    512
#define BATCH  64
#define VOCAB  32000
#define GATES  2048   // 4*HID
#define SRCLEN 64
#define ROWS   (SRCLEN * BATCH)   // 4096 encoder (t,n) rows

__device__ __forceinline__ v8f wmma_f32(v2f a, v2f b, v8f c) {
    // V_WMMA_F32_16X16X4_F32 : D = A(16x4) * B(4x16) + C(16x16), f32 throughout
    return __builtin_amdgcn_wmma_f32_16x16x4_f32(
        /*neg_a=*/false, a, /*neg_b=*/false, b,
        /*c_mod=*/(short)0, c, /*reuse_a=*/false, /*reuse_b=*/false);
}

__device__ __forceinline__ float sigm(float x) {
    return 1.0f / (1.0f + expf(-x));
}

// -------------------- generic fill --------------------
__global__ void fill_zero(float* __restrict__ p, long long n) {
    long long i = (long long)blockIdx.x * blockDim.x + threadIdx.x;
    if (i < n) p[i] = 0.0f;
}

// -------------------- batched input projection --------------------
// Z[r, :] = A[r, :] @ W^T + (bih + bhh),  A rows via optional token gather.
// grid (ROWS/16, 32), 4 waves/block; wave w owns gate-tile blockIdx.y*4+w.
__global__ __launch_bounds__(128) void gemm_zih_wmma(
    const int*   __restrict__ tokens,  // may be null
    const float* __restrict__ Xin,     // emb table (tokens!=null) or [ROWS,512]
    const float* __restrict__ W,       // [2048, 512]
    const float* __restrict__ bih,     // [2048]
    const float* __restrict__ bhh,     // [2048]
    float*       __restrict__ Z) {     // [ROWS, 2048]
    const int mt    = blockIdx.x;
    const int ntile = blockIdx.y * 4 + (threadIdx.x >> 5);
    const int lane  = threadIdx.x & 31;
    const int rl    = lane & 15;
    const int koff  = (lane & 16) ? 2 : 0;
    const int m0    = mt * 16;
    const int n0    = ntile * 16;

    const int row = m0 + rl;
    const float* xr = tokens ? (Xin + (size_t)tokens[row] * HID)
                             : (Xin + (size_t)row * HID);
    const float* wr = W + (size_t)(n0 + rl) * HID;

    v8f acc0 = {};
    v8f acc1 = {};
    for (int k = 0; k < HID; k += 8) {
        v2f a0 = *(const v2f*)(xr + k + koff);
        v2f b0 = *(const v2f*)(wr + k + koff);
        acc0 = wmma_f32(a0, b0, acc0);
        v2f a1 = *(const v2f*)(xr + k + 4 + koff);
        v2f b1 = *(const v2f*)(wr + k + 4 + koff);
        acc1 = wmma_f32(a1, b1, acc1);
    }
    v8f z = acc0 + acc1;
    const float bias = bih[n0 + rl] + bhh[n0 + rl];
    const int mh = (lane & 16) ? 8 : 0;
#pragma unroll
    for (int r = 0; r < 8; ++r)
        Z[(size_t)(m0 + mh + r) * GATES + n0 + rl] = z[r] + bias;
}

// -------------------- recurrent-only LSTM cell (Zih precomputed) ----------
// z = Zih_tile (loaded straight into the WMMA C accumulator) + h @ Whh^T.
// grid (4, 32); wave g = gate g.
__global__ __launch_bounds__(128) void lstm_cell_hh_wmma(
    const float* __restrict__ zih,   // [64, 2048] slice (bias already added)
    const float* __restrict__ h_in,  // [64, 512]
    const float* __restrict__ c_in,  // [64, 512]
    const float* __restrict__ Whh,   // [2048, 512]
    float* __restrict__ h_out,
    float* __restrict__ c_out) {
    const int mt   = blockIdx.x;
    const int nt   = blockIdx.y;
    const int g    = threadIdx.x >> 5;
    const int lane = threadIdx.x & 31;
    const int rl   = lane & 15;
    const int koff = (lane & 16) ? 2 : 0;
    const int m0   = mt * 16;
    const int w0   = g * HID + nt * 16;
    const int mh   = (lane & 16) ? 8 : 0;

    // preload Zih tile in C-layout as the initial accumulator
    v8f acc0;
    const float* zr = zih + (size_t)(m0 + mh) * GATES + w0 + rl;
#pragma unroll
    for (int r = 0; r < 8; ++r) acc0[r] = zr[(size_t)r * GATES];
    v8f acc1 = {};

    const float* hr = h_in + (size_t)(m0 + rl) * HID;
    const float* wh = Whh  + (size_t)(w0 + rl) * HID;
    for (int k = 0; k < HID; k += 8) {
        v2f a0 = *(const v2f*)(hr + k + koff);
        v2f b0 = *(const v2f*)(wh + k + koff);
        acc0 = wmma_f32(a0, b0, acc0);
        v2f a1 = *(const v2f*)(hr + k + 4 + koff);
        v2f b1 = *(const v2f*)(wh + k + 4 + koff);
        acc1 = wmma_f32(a1, b1, acc1);
    }
    v8f z = acc0 + acc1;

    __shared__ float zsh[4][16][16];
#pragma unroll
    for (int r = 0; r < 8; ++r) zsh[g][mh + r][rl] = z[r];
    __syncthreads();

    for (int e = threadIdx.x; e < 256; e += 128) {
        const int ml = e >> 4, nl = e & 15;
        const float zi = zsh[0][ml][nl];
        const float zf = zsh[1][ml][nl];
        const float zg = zsh[2][ml][nl];
        const float zo = zsh[3][ml][nl];
        const int m  = m0 + ml;
        const int nh = blockIdx.y * 16 + nl;
        const float co = c_in[(size_t)m * HID + nh];
        const float cn = sigm(zf) * co + sigm(zi) * tanhf(zg);
        const float hn = sigm(zo) * tanhf(cn);
        c_out[(size_t)m * HID + nh] = cn;
        h_out[(size_t)m * HID + nh] = hn;
    }
}

// -------------------- full LSTM cell (dual GEMM, fused embedding) ---------
// x rows come from xsrc directly, or gathered via tokens (embedding fused).
__global__ __launch_bounds__(128) void lstm_cell_wmma(
    const int*   __restrict__ tokens, // may be null
    const float* __restrict__ xsrc,   // emb table (tokens!=null) or [64,512]
    const float* __restrict__ h_in,
    const float* __restrict__ c_in,
    const float* __restrict__ Wih,
    const float* __restrict__ Whh,
    const float* __restrict__ bih,
    const float* __restrict__ bhh,
    float* __restrict__ h_out,
    float* __restrict__ c_out) {
    const int mt   = blockIdx.x;
    const int nt   = blockIdx.y;
    const int g    = threadIdx.x >> 5;
    const int lane = threadIdx.x & 31;
    const int rl   = lane & 15;
    const int koff = (lane & 16) ? 2 : 0;
    const int m0   = mt * 16;
    const int w0   = g * HID + nt * 16;

    const float* xr = tokens ? (xsrc + (size_t)tokens[m0 + rl] * HID)
                             : (xsrc + (size_t)(m0 + rl) * HID);
    const float* wi = Wih + (size_t)(w0 + rl) * HID;
    v8f acc0 = {};
    v8f acc1 = {};
    for (int k = 0; k < HID; k += 8) {
        v2f a0 = *(const v2f*)(xr + k + koff);
        v2f b0 = *(const v2f*)(wi + k + koff);
        acc0 = wmma_f32(a0, b0, acc0);
        v2f a1 = *(const v2f*)(xr + k + 4 + koff);
        v2f b1 = *(const v2f*)(wi + k + 4 + koff);
        acc1 = wmma_f32(a1, b1, acc1);
    }
    const float* hr = h_in + (size_t)(m0 + rl) * HID;
    const float* wh = Whh  + (size_t)(w0 + rl) * HID;
    for (int k = 0; k < HID; k += 8) {
        v2f a0 = *(const v2f*)(hr + k + koff);
        v2f b0 = *(const v2f*)(wh + k + koff);
        acc0 = wmma_f32(a0, b0, acc0);
        v2f a1 = *(const v2f*)(hr + k + 4 + koff);
        v2f b1 = *(const v2f*)(wh + k + 4 + koff);
        acc1 = wmma_f32(a1, b1, acc1);
    }
    v8f z = acc0 + acc1;
    const float bias = bih[w0 + rl] + bhh[w0 + rl];

    __shared__ float zsh[4][16][16];
    const int mh = (lane & 16) ? 8 : 0;
#pragma unroll
    for (int r = 0; r < 8; ++r) zsh[g][mh + r][rl] = z[r] + bias;
    __syncthreads();

    for (int e = threadIdx.x; e < 256; e += 128) {
        const int ml = e >> 4, nl = e & 15;
        const float zi = zsh[0][ml][nl];
        const float zf = zsh[1][ml][nl];
        const float zg = zsh[2][ml][nl];
        const float zo = zsh[3][ml][nl];
        const int m  = m0 + ml;
        const int nh = nt * 16 + nl;
        const float co = c_in[(size_t)m * HID + nh];
        const float cn = sigm(zf) * co + sigm(zi) * tanhf(zg);
        const float hn = sigm(zo) * tanhf(cn);
        c_out[(size_t)m * HID + nh] = cn;
        h_out[(size_t)m * HID + nh] = hn;
    }
}

// -------------------- FC / vocab projection --------------------
__global__ __launch_bounds__(128) void fc_wmma(
    const float* __restrict__ h,     // [64, 512]
    const float* __restrict__ W,     // [32000, 512]
    const float* __restrict__ b,     // [32000]
    float* __restrict__ out) {       // [64, 32000]
    const int mt    = blockIdx.x;
    const int ntile = blockIdx.y * 4 + (threadIdx.x >> 5);
    const int lane  = threadIdx.x & 31;
    const int rl    = lane & 15;
    const int koff  = (lane & 16) ? 2 : 0;
    const int m0    = mt * 16;
    const int n0    = ntile * 16;

    v8f acc0 = {};
    v8f acc1 = {};
    const float* hr = h + (size_t)(m0 + rl) * HID;
    const float* wr = W + (size_t)(n0 + rl) * HID;
    for (int k = 0; k < HID; k += 8) {
        v2f a0 = *(const v2f*)(hr + k + koff);
        v2f b0 = *(const v2f*)(wr + k + koff);
        acc0 = wmma_f32(a0, b0, acc0);
        v2f a1 = *(const v2f*)(hr + k + 4 + koff);
        v2f b1 = *(const v2f*)(wr + k + 4 + koff);
        acc1 = wmma_f32(a1, b1, acc1);
    }
    v8f z = acc0 + acc1;
    const float bias = b[n0 + rl];
    const int mh = (lane & 16) ? 8 : 0;
#pragma unroll
    for (int r = 0; r < 8; ++r)
        out[(size_t)(m0 + mh + r) * VOCAB + n0 + rl] = z[r] + bias;
}

// -------------------- row-wise argmax (first-max semantics) ---------------
__global__ void argmax_rows(const float* __restrict__ pred, int* __restrict__ tok) {
    const int row = blockIdx.x;
    const float* p = pred + (size_t)row * VOCAB;
    float best = -INFINITY;
    int   bi   = 0x7fffffff;
    for (int i = threadIdx.x; i < VOCAB; i += 256) {
        const float v = p[i];
        if (v > best || (v == best && i < bi)) { best = v; bi = i; }
    }
    __shared__ float sv[256];
    __shared__ int   si[256];
    sv[threadIdx.x] = best;
    si[threadIdx.x] = bi;
    __syncthreads();
    for (int s = 128; s > 0; s >>= 1) {
        if (threadIdx.x < s) {
            const float v2 = sv[threadIdx.x + s];
            const int   i2 = si[threadIdx.x + s];
            if (v2 > sv[threadIdx.x] ||
                (v2 == sv[threadIdx.x] && i2 < si[threadIdx.x])) {
                sv[threadIdx.x] = v2;
                si[threadIdx.x] = i2;
            }
        }
        __syncthreads();
    }
    if (threadIdx.x == 0) tok[row] = si[0];
}

extern "C" void kernel_launch(void* const* d_in, const int* in_sizes, int n_in,
                              void* d_out, int out_size, void* d_ws, size_t ws_size,
                              hipStream_t stream) {
    (void)in_sizes; (void)n_in; (void)out_size;
    const int*   src     = (const int*)  d_in[0];
    const int*   tgt     = (const int*)  d_in[1];
    const float* enc_emb = (const float*)d_in[2];
    const float* enc_Wih = (const float*)d_in[3];
    const float* enc_Whh = (const float*)d_in[4];
    const float* enc_bih = (const float*)d_in[5];
    const float* enc_bhh = (const float*)d_in[6];
    const float* dec_emb = (const float*)d_in[7];
    const float* dec_Wih = (const float*)d_in[8];
    const float* dec_Whh = (const float*)d_in[9];
    const float* dec_bih = (const float*)d_in[10];
    const float* dec_bhh = (const float*)d_in[11];
    const float* fc_W    = (const float*)d_in[12];
    const float* fc_b    = (const float*)d_in[13];
    float* out = (float*)d_out;

    const size_t NB = (size_t)BATCH * HID;   // 32768 floats
    float* ws      = (float*)d_ws;
    float* zerobuf = ws;                     // NB (read-only zero state)
    float* h0p[2]  = { ws + 1 * NB, ws + 2 * NB };
    float* c0p[2]  = { ws + 3 * NB, ws + 4 * NB };
    float* h1p[2]  = { ws + 5 * NB, ws + 6 * NB };
    float* c1p[2]  = { ws + 7 * NB, ws + 8 * NB };
    int*   ntok    = (int*)(ws + 9 * NB);
    float* zbuf    = ws + 9 * NB + 64;                    // [4096, 2048]
    float* h0_seq  = zbuf + (size_t)ROWS * GATES;         // [4096, 512]

    const size_t need_bytes =
        (9 * NB + 64 + (size_t)ROWS * GATES + (size_t)ROWS * HID) * sizeof(float);
    const bool batched = ws_size >= need_bytes;

    const int zgrid = (int)((NB + 255) / 256);
    fill_zero<<<zgrid, 256, 0, stream>>>(zerobuf, (long long)NB);

    const size_t LW = (size_t)GATES * HID;
    const dim3 cgrid(4, 32);

    const float* dh0; const float* dc0;   // decoder initial states
    const float* dh1; const float* dc1;

    if (batched) {
        // ---- encoder layer 0: one big ih-projection GEMM, then hh-only steps
        const dim3 ggrid(ROWS / 16, 32);
        gemm_zih_wmma<<<ggrid, 128, 0, stream>>>(src, enc_emb, enc_Wih,
                                                 enc_bih, enc_bhh, zbuf);
        const float* hp = zerobuf; const float* cp = zerobuf;
        for (int t = 0; t < SRCLEN; ++t) {
            float* hout = h0_seq + (size_t)t * NB;
            float* cout = c0p[t & 1];
            lstm_cell_hh_wmma<<<cgrid, 128, 0, stream>>>(
                zbuf + (size_t)t * BATCH * GATES, hp, cp, enc_Whh, hout, cout);
            hp = hout; cp = cout;
        }
        dh0 = hp; dc0 = cp;

        // ---- encoder layer 1: big GEMM over the layer-0 hidden sequence
        gemm_zih_wmma<<<ggrid, 128, 0, stream>>>(nullptr, h0_seq, enc_Wih + LW,
                                                 enc_bih + GATES, enc_bhh + GATES,
                                                 zbuf);
        hp = zerobuf; cp = zerobuf;
        for (int t = 0; t < SRCLEN; ++t) {
            float* hout = h1p[t & 1];
            float* cout = c1p[t & 1];
            lstm_cell_hh_wmma<<<cgrid, 128, 0, stream>>>(
                zbuf + (size_t)t * BATCH * GATES, hp, cp, enc_Whh + LW, hout, cout);
            hp = hout; cp = cout;
        }
        dh1 = hp; dc1 = cp;
    } else {
        // ---- fallback: per-step full cells (embedding fused into GEMM)
        const float* h0 = zerobuf; const float* c0 = zerobuf;
        const float* h1 = zerobuf; const float* c1 = zerobuf;
        for (int t = 0; t < SRCLEN; ++t) {
            const int q = t & 1;
            lstm_cell_wmma<<<cgrid, 128, 0, stream>>>(
                src + t * BATCH, enc_emb, h0, c0,
                enc_Wih, enc_Whh, enc_bih, enc_bhh, h0p[q], c0p[q]);
            h0 = h0p[q]; c0 = c0p[q];
            lstm_cell_wmma<<<cgrid, 128, 0, stream>>>(
                nullptr, h0, h1, c1,
                enc_Wih + LW, enc_Whh + LW, enc_bih + GATES, enc_bhh + GATES,
                h1p[q], c1p[q]);
            h1 = h1p[q]; c1 = c1p[q];
        }
        dh0 = h0; dc0 = c0; dh1 = h1; dc1 = c1;
    }

    // ---------------- decoder: 31 greedy steps ----------------
    const long long slice = (long long)BATCH * VOCAB;
    fill_zero<<<(int)((slice + 255) / 256), 256, 0, stream>>>(out, slice); // outputs[0]

    const dim3 fgrid(4, 500);
    for (int s = 0; s < 31; ++s) {
        const int q = s & 1;
        // guarantee write buffer != read buffer regardless of path history:
        float* w_h0 = (h0p[q] == dh0) ? h0p[q ^ 1] : h0p[q];
        float* w_c0 = (c0p[q] == dc0) ? c0p[q ^ 1] : c0p[q];
        float* w_h1 = (h1p[q] == dh1) ? h1p[q ^ 1] : h1p[q];
        float* w_c1 = (c1p[q] == dc1) ? c1p[q ^ 1] : c1p[q];

        const int* toks = (s == 0) ? tgt : ntok;   // tgt[0] seeds, then greedy
        lstm_cell_wmma<<<cgrid, 128, 0, stream>>>(
            toks, dec_emb, dh0, dc0,
            dec_Wih, dec_Whh, dec_bih, dec_bhh, w_h0, w_c0);
        dh0 = w_h0; dc0 = w_c0;

        lstm_cell_wmma<<<cgrid, 128, 0, stream>>>(
            nullptr, dh0, dh1, dc1,
            dec_Wih + LW, dec_Whh + LW, dec_bih + GATES, dec_bhh + GATES,
            w_h1, w_c1);
        dh1 = w_h1; dc1 = w_c1;

        float* pslice = out + (long long)(s + 1) * slice;
        fc_wmma<<<fgrid, 128, 0, stream>>>(dh1, fc_W, fc_b, pslice);
        argmax_rows<<<64, 256, 0, stream>>>(pslice, ntok);
    }
}